// Kwta_87522843560186
// MI455X (gfx1250) — compile-verified
//
#include <hip/hip_runtime.h>
#include <hip/hip_bf16.h>
#include <stdint.h>

// ---------------- problem constants (match reference) ----------------
#define CH    256          // channels per row
#define KSEL  26           // k = max(1, round(0.1 * 256))
#define WPB   8            // waves per block (wave32 => 256 threads)
#define BLOCK (WPB * 32)

// s_wait_asynccnt <= n  (CDNA5 ASYNCcnt; also acts as compiler memory barrier)
#define WAIT_ASYNCCNT(n) asm volatile("s_wait_asynccnt " #n ::: "memory")

static __device__ __forceinline__ uint32_t umin32(uint32_t a, uint32_t b) {
  return a < b ? a : b;
}

// Monotone bijection f32 -> u32 (total order matches float compare for non-NaN)
static __device__ __forceinline__ uint32_t f2k(float f) {
  uint32_t u = __float_as_uint(f);
  return u ^ ((uint32_t)((int32_t)u >> 31) | 0x80000000u);
}

// Stage one 1KB row (256 f32) from global into LDS with the CDNA5 async copy
// engine: 2 x global_load_async_to_lds_b128, 16B per lane per instruction.
static __device__ __forceinline__ void async_load_row(const float* g,
                                                      const float* l,
                                                      int lane) {
  uint64_t ga = (uint64_t)(uintptr_t)(g + (lane << 2));
  // generic LDS pointer: low 32 bits are the workgroup-relative LDS byte offset
  uint32_t la = (uint32_t)(uintptr_t)(l + (lane << 2));
  asm volatile("global_load_async_to_lds_b128 %0, %1, off"
               :: "v"(la), "v"(ga) : "memory");
  asm volatile("global_load_async_to_lds_b128 %0, %1, off"
               :: "v"(la + 512u), "v"(ga + 512ull) : "memory");
}

__global__ __launch_bounds__(BLOCK, 1)
void kwta_kernel(const float* __restrict__ x, float* __restrict__ out, int nrows) {
  __shared__ __align__(16) float stage[WPB][2][CH];   // 16 KB: double buffer per wave

  const int lane = (int)(threadIdx.x & 31u);
  const int w    = (int)(threadIdx.x >> 5);
  const long long wave0 = (long long)blockIdx.x * WPB + w;
  const long long nw    = (long long)gridDim.x * WPB;

  long long row = wave0;
  if (row < nrows) async_load_row(x + row * CH, &stage[w][0][0], lane);

  int buf = 0;
  for (; row < nrows; row += nw) {
    const long long nxt = row + nw;
    const bool pre = (nxt < nrows);
    if (pre) {
      async_load_row(x + nxt * CH, &stage[w][buf ^ 1][0], lane);
      WAIT_ASYNCCNT(2);      // 2 prefetch loads may remain in flight
    } else {
      WAIT_ASYNCCNT(0);
    }

    const float* rb = &stage[w][buf][0];
    const float4 a = *(const float4*)(rb + 8 * lane);
    const float4 b = *(const float4*)(rb + 8 * lane + 4);

    const uint32_t k0 = f2k(a.x), k1 = f2k(a.y), k2 = f2k(a.z), k3 = f2k(a.w);
    const uint32_t k4 = f2k(b.x), k5 = f2k(b.y), k6 = f2k(b.z), k7 = f2k(b.w);

    // Radix descent on key bits. Invariant: count(>=P) >= KSEL > count(>=P+2^(bit+1)).
    // Wave-wide counts via v_cmp->VCC ballot + scalar popcount (8 VALU/iter).
    uint32_t P = 0u;
#pragma unroll 1
    for (int bit = 31; bit >= 0; --bit) {
      const uint32_t T = P | (1u << bit);
      int c;
      c  = __builtin_popcount(__builtin_amdgcn_ballot_w32(k0 >= T));
      c += __builtin_popcount(__builtin_amdgcn_ballot_w32(k1 >= T));
      c += __builtin_popcount(__builtin_amdgcn_ballot_w32(k2 >= T));
      c += __builtin_popcount(__builtin_amdgcn_ballot_w32(k3 >= T));
      c += __builtin_popcount(__builtin_amdgcn_ballot_w32(k4 >= T));
      c += __builtin_popcount(__builtin_amdgcn_ballot_w32(k5 >= T));
      c += __builtin_popcount(__builtin_amdgcn_ballot_w32(k6 >= T));
      c += __builtin_popcount(__builtin_amdgcn_ballot_w32(k7 >= T));
      if (c >= KSEL) {
        P = T;
        if (c == KSEL) break;   // exactly K keys >= P: threshold = min{key >= P}
      }
    }

    // threshold key = min over keys >= P (valid for both early and full exit)
    uint32_t m;
    m = (k0 >= P) ? k0 : 0xFFFFFFFFu;
    m = umin32(m, (k1 >= P) ? k1 : 0xFFFFFFFFu);
    m = umin32(m, (k2 >= P) ? k2 : 0xFFFFFFFFu);
    m = umin32(m, (k3 >= P) ? k3 : 0xFFFFFFFFu);
    m = umin32(m, (k4 >= P) ? k4 : 0xFFFFFFFFu);
    m = umin32(m, (k5 >= P) ? k5 : 0xFFFFFFFFu);
    m = umin32(m, (k6 >= P) ? k6 : 0xFFFFFFFFu);
    m = umin32(m, (k7 >= P) ? k7 : 0xFFFFFFFFu);
#pragma unroll
    for (int off = 16; off > 0; off >>= 1) {
      uint32_t o = (uint32_t)__shfl_xor((int)m, off, 32);
      m = umin32(m, o);
    }
    const uint32_t thr = m;

    float4 oa, ob;
    oa.x = (k0 >= thr) ? a.x : 0.0f;
    oa.y = (k1 >= thr) ? a.y : 0.0f;
    oa.z = (k2 >= thr) ? a.z : 0.0f;
    oa.w = (k3 >= thr) ? a.w : 0.0f;
    ob.x = (k4 >= thr) ? b.x : 0.0f;
    ob.y = (k5 >= thr) ? b.y : 0.0f;
    ob.z = (k6 >= thr) ? b.z : 0.0f;
    ob.w = (k7 >= thr) ? b.w : 0.0f;

    float* orow = out + row * CH;
    *(float4*)(orow + 8 * lane)     = oa;
    *(float4*)(orow + 8 * lane + 4) = ob;

    buf ^= 1;
  }
}

extern "C" void kernel_launch(void* const* d_in, const int* in_sizes, int n_in,
                              void* d_out, int out_size, void* d_ws, size_t ws_size,
                              hipStream_t stream) {
  (void)n_in; (void)out_size; (void)d_ws; (void)ws_size;
  const float* x = (const float*)d_in[0];
  float* out = (float*)d_out;
  const int nrows = in_sizes[0] / CH;   // 200704 for the reference shapes

  int blocks = 1024;                    // 8192 waves, grid-stride over rows
  int maxb = (nrows + WPB - 1) / WPB;
  if (blocks > maxb) blocks = maxb;
  if (blocks < 1) blocks = 1;

  kwta_kernel<<<blocks, BLOCK, 0, stream>>>(x, out, nrows);
}